// AutoCorrelation_7851200217094
// MI455X (gfx1250) — compile-verified
//
#include <hip/hip_runtime.h>
#include <math.h>

// AutoCorrelation (Autoformer) for B=8, L=2048, H=8, D=64 on gfx1250.
// Phase 1: circular cross-correlation via V_WMMA_F32_16X16X4_F32, accumulated
//          along tile-diagonals so the L x L product matrix never materializes.
// Phase 2: per-batch top-7 + softmax (wave32 butterfly argmax).
// Phase 3: 7-tap weighted gather of v (repeat_interleave(2) semantics).

typedef __attribute__((ext_vector_type(2))) float v2f;
typedef __attribute__((ext_vector_type(8))) float v8f;

#define LB   8
#define LL   2048
#define LMSK (LL - 1)
#define CH   512      // H*D
#define TOPK 7

// ---------------------------------------------------------------- zero scratch
__global__ void zero_corr(float* __restrict__ corr) {
  int i = blockIdx.x * blockDim.x + threadIdx.x;
  if (i < LB * LL) corr[i] = 0.0f;
}

// ------------------------------------------------------- phase 1: WMMA corr
// grid = (32, 8): blockIdx.x = group of 4 diagonal tiles (64 tau values),
//                 blockIdx.y = batch.  block = 128 threads = 4 waves.
// Each wave sums 32 j-tiles along the diagonal; 4 f32 accumulators reuse one
// B-matrix load.  Tau binning: tau = delta_base + d*16 + (M - N), M-N in
// [-15,15] -> 79 bins reduced in LDS, then ~79 global atomics per workgroup.
__global__ __launch_bounds__(128) void corr_wmma(const float* __restrict__ q,
                                                 const float* __restrict__ k,
                                                 float* __restrict__ corr) {
  __shared__ float bins[80];
  const int b          = blockIdx.y;
  const int delta_base = blockIdx.x * 64;      // 4 tile-diagonals of 16
  const int wave = threadIdx.x >> 5;
  const int lane = threadIdx.x & 31;
  const int half = lane >> 4;                  // 0: K pair {0,1}, 1: {2,3}
  const int lrow = lane & 15;                  // M (A) / N (B) within tile

  for (int i = threadIdx.x; i < 80; i += 128) bins[i] = 0.0f;
  __syncthreads();

  const float* __restrict__ qb = q + (size_t)b * LL * CH;
  const float* __restrict__ kb = k + (size_t)b * LL * CH;

  v8f acc0 = {}, acc1 = {}, acc2 = {}, acc3 = {};

  // Each wave handles j-tiles [wave*32, wave*32+32)
  for (int jt = wave * 32; jt < wave * 32 + 32; ++jt) {
    const int j0 = jt * 16;
    // B operand: k rows j0+lrow, channel pair selected by half.
    const float* bp  = kb + (size_t)(j0 + lrow) * CH + half * 2;
    // A operands: q rows shifted by the 4 diagonal offsets (mod L).
    const float* ap0 = qb + (size_t)((j0 + delta_base +  0 + lrow) & LMSK) * CH + half * 2;
    const float* ap1 = qb + (size_t)((j0 + delta_base + 16 + lrow) & LMSK) * CH + half * 2;
    const float* ap2 = qb + (size_t)((j0 + delta_base + 32 + lrow) & LMSK) * CH + half * 2;
    const float* ap3 = qb + (size_t)((j0 + delta_base + 48 + lrow) & LMSK) * CH + half * 2;

    for (int c0 = 0; c0 < CH; c0 += 4) {
      v2f bv = *(const v2f*)(bp  + c0);
      v2f a0 = *(const v2f*)(ap0 + c0);
      v2f a1 = *(const v2f*)(ap1 + c0);
      v2f a2 = *(const v2f*)(ap2 + c0);
      v2f a3 = *(const v2f*)(ap3 + c0);
      // D = A x B^T-style correlation step, K=4 channels per instruction.
      acc0 = __builtin_amdgcn_wmma_f32_16x16x4_f32(false, a0, false, bv, (short)0, acc0, false, false);
      acc1 = __builtin_amdgcn_wmma_f32_16x16x4_f32(false, a1, false, bv, (short)0, acc1, false, false);
      acc2 = __builtin_amdgcn_wmma_f32_16x16x4_f32(false, a2, false, bv, (short)0, acc2, false, false);
      acc3 = __builtin_amdgcn_wmma_f32_16x16x4_f32(false, a3, false, bv, (short)0, acc3, false, false);
    }
  }

  // C/D layout: VGPR r, lanes 0-15 -> (M=r, N=lane); lanes 16-31 -> (M=r+8).
  const int N = lrow;
  #pragma unroll
  for (int r = 0; r < 8; ++r) {
    const int mn = (r + (half << 3)) - N;      // in [-15, 15]
    atomicAdd(&bins[mn + 15 +  0], acc0[r]);
    atomicAdd(&bins[mn + 15 + 16], acc1[r]);
    atomicAdd(&bins[mn + 15 + 32], acc2[r]);
    atomicAdd(&bins[mn + 15 + 48], acc3[r]);
  }
  __syncthreads();

  if (threadIdx.x < 79) {
    const int tau = (delta_base + (int)threadIdx.x - 15) & LMSK;
    atomicAdd(&corr[b * LL + tau], bins[threadIdx.x] * (1.0f / 512.0f));
  }
}

// ------------------------------------------------ phase 2: top-7 + softmax
// One wave32 per batch.  Iterative argmax with lower-index tie-break (matches
// lax.top_k ordering), then softmax on lane 0.
__global__ __launch_bounds__(32) void topk_softmax(const float* __restrict__ corr,
                                                   float* __restrict__ alpha,
                                                   int* __restrict__ delay) {
  const int b    = blockIdx.x;
  const int lane = threadIdx.x;            // 0..31
  const float* c = corr + b * LL;

  float vals[LL / 32];
  #pragma unroll
  for (int i = 0; i < LL / 32; ++i) vals[i] = c[lane + i * 32];

  float chosenVal[TOPK];
  int   chosenIdx[TOPK];

  for (int it = 0; it < TOPK; ++it) {
    float best = -3.402823466e+38f;
    int   bidx = 0x7fffffff;
    #pragma unroll
    for (int i = 0; i < LL / 32; ++i) {
      const int   idx = lane + i * 32;
      const float v   = vals[i];
      if (v > best || (v == best && idx < bidx)) { best = v; bidx = idx; }
    }
    // wave32 butterfly reduction
    #pragma unroll
    for (int off = 16; off > 0; off >>= 1) {
      const float ov = __shfl_xor(best, off, 32);
      const int   oi = __shfl_xor(bidx, off, 32);
      if (ov > best || (ov == best && oi < bidx)) { best = ov; bidx = oi; }
    }
    chosenVal[it] = best;
    chosenIdx[it] = bidx;
    if ((bidx & 31) == lane) vals[bidx >> 5] = -3.402823466e+38f;  // remove winner
  }

  if (lane == 0) {
    const float m = chosenVal[0];            // sorted descending -> max first
    float e[TOPK], s = 0.0f;
    #pragma unroll
    for (int i = 0; i < TOPK; ++i) { e[i] = __expf(chosenVal[i] - m); s += e[i]; }
    const float inv = 1.0f / s;
    #pragma unroll
    for (int i = 0; i < TOPK; ++i) {
      alpha[b * TOPK + i] = e[i] * inv;
      delay[b * TOPK + i] = chosenIdx[i];
    }
  }
}

// -------------------------------------------------- phase 3: weighted gather
// out[b,t,h,d] = sum_i alpha[b,i] * v[b, (t+delay[b,i])>>1, h, d]
// grid = B*L blocks, 128 threads, float4 per thread over the 512 channels.
__global__ __launch_bounds__(128) void gather_out(const float* __restrict__ v,
                                                  const float* __restrict__ alpha,
                                                  const int* __restrict__ delay,
                                                  float* __restrict__ out) {
  const int bt = blockIdx.x;                 // b*L + t
  const int b  = bt >> 11;
  const int t  = bt & LMSK;
  const int c4 = threadIdx.x;                // channel group, 4 floats each

  float a[TOPK];
  int   dl[TOPK];
  #pragma unroll
  for (int i = 0; i < TOPK; ++i) {
    a[i]  = alpha[b * TOPK + i];
    dl[i] = delay[b * TOPK + i];
  }

  const float* __restrict__ vb = v + (size_t)b * LL * CH;
  float4 acc = make_float4(0.f, 0.f, 0.f, 0.f);
  #pragma unroll
  for (int i = 0; i < TOPK; ++i) {
    const int row = (t + dl[i]) >> 1;        // repeat_interleave(2) gather
    const float4 g = *(const float4*)(vb + (size_t)row * CH + c4 * 4);
    acc.x += a[i] * g.x;
    acc.y += a[i] * g.y;
    acc.z += a[i] * g.z;
    acc.w += a[i] * g.w;
  }
  *(float4*)(out + (size_t)bt * CH + c4 * 4) = acc;
}

// --------------------------------------------------------------------- launch
extern "C" void kernel_launch(void* const* d_in, const int* in_sizes, int n_in,
                              void* d_out, int out_size, void* d_ws, size_t ws_size,
                              hipStream_t stream) {
  const float* q = (const float*)d_in[0];
  const float* k = (const float*)d_in[1];
  const float* v = (const float*)d_in[2];
  float* out = (float*)d_out;

  // workspace: corr[8][2048] | alpha[8][7] | delay[8][7]  (~66 KB)
  float* corr  = (float*)d_ws;
  float* alpha = corr + LB * LL;
  int*   delay = (int*)(alpha + LB * TOPK);

  zero_corr   <<<dim3((LB * LL + 255) / 256), dim3(256), 0, stream>>>(corr);
  corr_wmma   <<<dim3(32, LB),                dim3(128), 0, stream>>>(q, k, corr);
  topk_softmax<<<dim3(LB),                    dim3(32),  0, stream>>>(corr, alpha, delay);
  gather_out  <<<dim3(LB * LL),               dim3(128), 0, stream>>>(v, alpha, delay, out);
}